// TGCNCell_57406532878648
// MI455X (gfx1250) — compile-verified
//
#include <hip/hip_runtime.h>
#include <math.h>

// ---------------------------------------------------------------------------
// CDNA5 fp32 WMMA: D(16x16,f32) = A(16x4,f32) x B(4x16,f32) + C
// A/B frags: 2 VGPRs/lane (float2), C/D: 8 VGPRs/lane.
// ---------------------------------------------------------------------------
typedef __attribute__((ext_vector_type(2))) float v2f;
typedef __attribute__((ext_vector_type(8))) float v8f;

#define HIDF 128
#define TILE_M 16

__device__ __forceinline__ v8f wmma4(v2f a, v2f b, v8f c) {
  // (neg_a, A, neg_b, B, c_mod, C, reuse_a, reuse_b)
  return __builtin_amdgcn_wmma_f32_16x16x4_f32(false, a, false, b, (short)0, c,
                                               false, false);
}

// ---------------------------------------------------------------------------
// gfx1250 async global->LDS copy (ASYNCcnt path, no VGPR round-trip).
// LDS dest address = low 32 bits of the generic pointer (LDS aperture
// places the byte offset in addr[31:0]).
// ---------------------------------------------------------------------------
__device__ __forceinline__ void async_copy16(const void* gptr, void* lptr) {
  unsigned lds = (unsigned)(uintptr_t)lptr;
  asm volatile("global_load_async_to_lds_b128 %0, %1, off"
               :
               : "v"(lds), "v"(gptr)
               : "memory");
}
__device__ __forceinline__ void async_wait0() {
  asm volatile("s_wait_asynccnt 0" ::: "memory");
}

// ---------------------------------------------------------------------------
// degree / normalization helpers
// ---------------------------------------------------------------------------
__global__ void k_init_deg(float* deg, int n) {
  int i = blockIdx.x * blockDim.x + threadIdx.x;
  if (i < n) deg[i] = 1.0f;  // self-loop weight
}

__global__ void k_deg_accum(const int* __restrict__ dst,
                            const float* __restrict__ ew, float* deg, int nE) {
  int e = blockIdx.x * blockDim.x + threadIdx.x;
  if (e < nE) unsafeAtomicAdd(&deg[dst[e]], ew[e]);
}

__global__ void k_dinv(float* deg, int n) {
  int i = blockIdx.x * blockDim.x + threadIdx.x;
  if (i < n) {
    float d = deg[i];
    deg[i] = (d > 0.0f) ? rsqrtf(d) : 0.0f;
  }
}

__global__ void k_zero(float4* p, int n4) {
  int i = blockIdx.x * blockDim.x + threadIdx.x;
  if (i < n4) p[i] = make_float4(0.f, 0.f, 0.f, 0.f);
}

// ---------------------------------------------------------------------------
// C[N,128] = A[N,128] @ W[128,128]  (fp32 WMMA, 16-row tile per block,
// 8 wave32s each owning a 16-col stripe; A tile async-staged in LDS)
// ---------------------------------------------------------------------------
__global__ __launch_bounds__(256) void k_gemm128(const float* __restrict__ A,
                                                 const float* __restrict__ W,
                                                 float* __restrict__ C) {
  __shared__ __attribute__((aligned(16))) float As[TILE_M * HIDF];
  const int row0 = blockIdx.x * TILE_M;
  const int tid = threadIdx.x;

  // 16*128 floats = 512 x b128; 256 threads -> 2 async copies each
  for (int i = tid; i < 512; i += 256) {
    int r = i >> 5, c4 = i & 31;
    async_copy16(A + (size_t)(row0 + r) * HIDF + c4 * 4, As + i * 4);
  }
  async_wait0();
  __syncthreads();

  const int lane = tid & 31;
  const int wave = tid >> 5;
  const int col = wave * 16 + (lane & 15);
  const int khalf = (lane >> 4) * 2;  // lanes 16-31 carry K+2,K+3
  const int arow = lane & 15;

  v8f acc = {};
  for (int k0 = 0; k0 < HIDF; k0 += 4) {
    v2f a, b;
    a.x = As[arow * HIDF + k0 + khalf];
    a.y = As[arow * HIDF + k0 + khalf + 1];
    b.x = W[(size_t)(k0 + khalf) * HIDF + col];
    b.y = W[(size_t)(k0 + khalf + 1) * HIDF + col];
    acc = wmma4(a, b, acc);
  }
  const int rbase = row0 + (lane >> 4) * 8;
  for (int v = 0; v < 8; ++v) C[(size_t)(rbase + v) * HIDF + col] = acc[v];
}

// ---------------------------------------------------------------------------
// scatter-add aggregation: one wave32 per edge, float4 per lane
// ---------------------------------------------------------------------------
__global__ __launch_bounds__(256) void k_edge_scatter(
    const int* __restrict__ src, const int* __restrict__ dst,
    const float* __restrict__ ew, const float* __restrict__ dinv,
    const float* __restrict__ feat, float* __restrict__ agg, int nE) {
  int e = blockIdx.x * 8 + (threadIdx.x >> 5);
  if (e >= nE) return;
  int lane = threadIdx.x & 31;
  int s = src[e], d = dst[e];
  float nrm = dinv[s] * ew[e] * dinv[d];
  float4 f = ((const float4*)(feat + (size_t)s * HIDF))[lane];
  float* p = agg + (size_t)d * HIDF + lane * 4;
  unsafeAtomicAdd(p + 0, f.x * nrm);
  unsafeAtomicAdd(p + 1, f.y * nrm);
  unsafeAtomicAdd(p + 2, f.z * nrm);
  unsafeAtomicAdd(p + 3, f.w * nrm);
}

// out = act(agg + feat*dinv^2 + bias); mode 0 = ReLU, 1 = sigmoid.
// (feat*dinv^2 is the folded self-loop contribution)
__global__ void k_post(const float* __restrict__ agg,
                       const float* __restrict__ feat,
                       const float* __restrict__ dinv,
                       const float* __restrict__ bias, float* __restrict__ out,
                       int nElem, int mode) {
  int idx = blockIdx.x * blockDim.x + threadIdx.x;
  if (idx >= nElem) return;
  int i = idx >> 7, f = idx & (HIDF - 1);
  float dv = dinv[i];
  float val = agg[idx] + feat[idx] * dv * dv + bias[f];
  out[idx] = mode ? (1.0f / (1.0f + __expf(-val))) : fmaxf(val, 0.0f);
}

// ---------------------------------------------------------------------------
// Fused GRU gate: 16 nodes per block, 8 wave32s.
//   u = sigmoid([xg|h]@Wu + bu); r = sigmoid([xg|h]@Wr + br)
//   c = tanh([xg|r*h]@Wc + bc);  out = u*h + (1-u)*c
// ---------------------------------------------------------------------------
__global__ __launch_bounds__(256) void k_gru(
    const float* __restrict__ xg, const float* __restrict__ h,
    const float* __restrict__ Wu, const float* __restrict__ bu,
    const float* __restrict__ Wr, const float* __restrict__ br,
    const float* __restrict__ Wc, const float* __restrict__ bc,
    float* __restrict__ out) {
  __shared__ __attribute__((aligned(16))) float xs[TILE_M * HIDF];
  __shared__ __attribute__((aligned(16))) float hs[TILE_M * HIDF];
  __shared__ __attribute__((aligned(16))) float rh[TILE_M * HIDF];

  const int node0 = blockIdx.x * TILE_M;
  const int tid = threadIdx.x;
  for (int i = tid; i < 512; i += 256) {
    int r = i >> 5, c4 = i & 31;
    async_copy16(xg + (size_t)(node0 + r) * HIDF + c4 * 4, xs + i * 4);
    async_copy16(h + (size_t)(node0 + r) * HIDF + c4 * 4, hs + i * 4);
  }
  async_wait0();
  __syncthreads();

  const int lane = tid & 31;
  const int wave = tid >> 5;
  const int col = wave * 16 + (lane & 15);
  const int khalf = (lane >> 4) * 2;
  const int arow = lane & 15;
  const int rbase = (lane >> 4) * 8;

  // ---- u and r: K = 256 (xg half then h half) ----
  v8f uacc = {}, racc = {};
  for (int k0 = 0; k0 < 2 * HIDF; k0 += 4) {
    const float* S = (k0 < HIDF) ? xs : hs;
    int kk = (k0 & (HIDF - 1)) + khalf;
    v2f a, b0, b1;
    a.x = S[arow * HIDF + kk];
    a.y = S[arow * HIDF + kk + 1];
    b0.x = Wu[(size_t)(k0 + khalf) * HIDF + col];
    b0.y = Wu[(size_t)(k0 + khalf + 1) * HIDF + col];
    b1.x = Wr[(size_t)(k0 + khalf) * HIDF + col];
    b1.y = Wr[(size_t)(k0 + khalf + 1) * HIDF + col];
    uacc = wmma4(a, b0, uacc);
    racc = wmma4(a, b1, racc);
  }

  float u[8];
  for (int v = 0; v < 8; ++v) {
    u[v] = 1.0f / (1.0f + __expf(-(uacc[v] + bu[col])));
    float rv = 1.0f / (1.0f + __expf(-(racc[v] + br[col])));
    rh[(rbase + v) * HIDF + col] = rv * hs[(rbase + v) * HIDF + col];
  }
  __syncthreads();

  // ---- c: K = 256 (xg half then r*h half) ----
  v8f cacc = {};
  for (int k0 = 0; k0 < 2 * HIDF; k0 += 4) {
    const float* S = (k0 < HIDF) ? xs : rh;
    int kk = (k0 & (HIDF - 1)) + khalf;
    v2f a, b;
    a.x = S[arow * HIDF + kk];
    a.y = S[arow * HIDF + kk + 1];
    b.x = Wc[(size_t)(k0 + khalf) * HIDF + col];
    b.y = Wc[(size_t)(k0 + khalf + 1) * HIDF + col];
    cacc = wmma4(a, b, cacc);
  }
  for (int v = 0; v < 8; ++v) {
    float c = tanhf(cacc[v] + bc[col]);
    float hv = hs[(rbase + v) * HIDF + col];
    out[(size_t)(node0 + rbase + v) * HIDF + col] = u[v] * hv + (1.0f - u[v]) * c;
  }
}

// ---------------------------------------------------------------------------
extern "C" void kernel_launch(void* const* d_in, const int* in_sizes, int n_in,
                              void* d_out, int out_size, void* d_ws,
                              size_t ws_size, hipStream_t stream) {
  const float* x = (const float*)d_in[0];
  const int* ei = (const int*)d_in[1];  // [2, E]: row0 = src, row1 = dst
  const float* ew = (const float*)d_in[2];
  const float* h = (const float*)d_in[3];
  const float* W1 = (const float*)d_in[4];
  const float* b1 = (const float*)d_in[5];
  const float* W2 = (const float*)d_in[6];
  const float* b2 = (const float*)d_in[7];
  const float* Wu = (const float*)d_in[8];
  const float* bu = (const float*)d_in[9];
  const float* Wr = (const float*)d_in[10];
  const float* br = (const float*)d_in[11];
  const float* Wc = (const float*)d_in[12];
  const float* bc = (const float*)d_in[13];

  const int N = in_sizes[0] / HIDF;  // 20000
  const int E = in_sizes[2];         // 640000
  const int nElem = N * HIDF;
  const int* src = ei;
  const int* dst = ei + E;

  float* ws = (float*)d_ws;
  float* dinv = ws;                              // N floats (deg -> dinv)
  float* bufA = ws + ((N + 255) & ~255);         // N*128 (xw1 / hw2)
  float* bufB = bufA + (size_t)N * HIDF;         // N*128 (agg, zeroed)
  float* bufC = bufB + (size_t)N * HIDF;         // N*128 (h1 / xg)

  const int T = 256;
  dim3 blk(T);
  dim3 gN((N + T - 1) / T), gE((E + T - 1) / T), gElem((nElem + T - 1) / T);
  dim3 gZero((nElem / 4 + T - 1) / T);
  dim3 gTile(N / TILE_M);           // N divisible by 16
  dim3 gEdge((E + 7) / 8);          // 8 edges (waves) per block

  // normalization
  k_init_deg<<<gN, blk, 0, stream>>>(dinv, N);
  k_deg_accum<<<gE, blk, 0, stream>>>(dst, ew, dinv, E);
  k_dinv<<<gN, blk, 0, stream>>>(dinv, N);

  // layer 1: xw = x@W1 ; agg ; h1 = relu(agg + self + b1)
  k_gemm128<<<gTile, blk, 0, stream>>>(x, W1, bufA);
  k_zero<<<gZero, blk, 0, stream>>>((float4*)bufB, nElem / 4);
  k_edge_scatter<<<gEdge, blk, 0, stream>>>(src, dst, ew, dinv, bufA, bufB, E);
  k_post<<<gElem, blk, 0, stream>>>(bufB, bufA, dinv, b1, bufC, nElem, 0);

  // layer 2: hw = h1@W2 ; agg ; xg = sigmoid(agg + self + b2)
  k_gemm128<<<gTile, blk, 0, stream>>>(bufC, W2, bufA);
  k_zero<<<gZero, blk, 0, stream>>>((float4*)bufB, nElem / 4);
  k_edge_scatter<<<gEdge, blk, 0, stream>>>(src, dst, ew, dinv, bufA, bufB, E);
  k_post<<<gElem, blk, 0, stream>>>(bufB, bufA, dinv, b2, bufC, nElem, 1);

  // fused GRU gate
  k_gru<<<gTile, blk, 0, stream>>>(bufC, h, Wu, bu, Wr, br, Wc, bc,
                                   (float*)d_out);
}